// SparseCodebookMoE_67010079752231
// MI455X (gfx1250) — compile-verified
//
#include <hip/hip_runtime.h>
#include <hip/hip_bf16.h>
#include <math.h>

#define BB   16384
#define EE   16
#define HH   256
#define CC   64
#define IN   320      // H + C
#define O4   1024     // 4H
#define MT   32       // tokens per block (two 16-row WMMA M-tiles)
#define XS   324      // x LDS stride (320 + 4, %64==4 -> conflict-free)
#define CH   128      // hid chunk columns
#define HS   132      // hid-chunk LDS stride (128 + 4)

typedef float v2f __attribute__((ext_vector_type(2)));
typedef float v8f __attribute__((ext_vector_type(8)));

__device__ __forceinline__ v8f wmma_f32(v2f a, v2f b, v8f c) {
  // V_WMMA_F32_16X16X4_F32: D = A(16x4,f32) * B(4x16,f32) + C(16x16,f32)
  return __builtin_amdgcn_wmma_f32_16x16x4_f32(false, a, false, b, (short)0, c,
                                               false, false);
}

// ---------------------------------------------------------------------------
// Kernel 1: router — cosine sim, gumbel softmax, top-2 mask, expert bucketing
// ---------------------------------------------------------------------------
__global__ __launch_bounds__(256) void router_kernel(
    const float* __restrict__ ce, const float* __restrict__ anchor,
    const float* __restrict__ gumbel, float* __restrict__ w_masked,
    int* __restrict__ cnts, int* __restrict__ tokList, float* __restrict__ wtList) {
  __shared__ float an[EE * CC];
  __shared__ float rinv[EE];
  const int tid = threadIdx.x;

  for (int i = tid; i < EE * CC; i += 256) an[i] = anchor[i];
  __syncthreads();
  if (tid < EE) {
    float s = 0.f;
    for (int c = 0; c < CC; ++c) { float v = an[tid * CC + c]; s += v * v; }
    rinv[tid] = 1.0f / fmaxf(sqrtf(s), 1e-8f);
  }
  __syncthreads();
  for (int i = tid; i < EE * CC; i += 256) an[i] *= rinv[i >> 6];
  __syncthreads();

  const int b = blockIdx.x * 256 + tid;

  float4 cv[CC / 4];
  const float4* cp = (const float4*)(ce + (size_t)b * CC);
  float ss = 0.f;
#pragma unroll
  for (int q = 0; q < CC / 4; ++q) {
    cv[q] = cp[q];
    ss += cv[q].x * cv[q].x + cv[q].y * cv[q].y + cv[q].z * cv[q].z + cv[q].w * cv[q].w;
  }
  const float ri = 1.0f / fmaxf(sqrtf(ss), 1e-8f);

  float w[EE];
  float zmax = -3.402823466e38f;
#pragma unroll
  for (int e = 0; e < EE; ++e) {
    const float4* ap = (const float4*)(an + e * CC);
    float d = 0.f;
#pragma unroll
    for (int q = 0; q < CC / 4; ++q) {
      float4 a4 = ap[q];
      d += cv[q].x * a4.x + cv[q].y * a4.y + cv[q].z * a4.z + cv[q].w * a4.w;
    }
    float logit = (d * ri) * 0.125f;
    float z = (logit + gumbel[(size_t)b * EE + e]) * 2.0f;  // /TAU, TAU=0.5
    w[e] = z;
    zmax = fmaxf(zmax, z);
  }
  float sum = 0.f;
#pragma unroll
  for (int e = 0; e < EE; ++e) { float p = expf(w[e] - zmax); w[e] = p; sum += p; }
  const float inv = 1.0f / sum;
#pragma unroll
  for (int e = 0; e < EE; ++e) w[e] *= inv;

  // top-2 (first-index-on-ties, matching lax.top_k)
  int i0 = 0; float v0 = w[0];
#pragma unroll
  for (int e = 1; e < EE; ++e) { if (w[e] > v0) { v0 = w[e]; i0 = e; } }
  int i1 = (i0 == 0) ? 1 : 0; float v1 = w[i1];
#pragma unroll
  for (int e = 0; e < EE; ++e) { if (e != i0 && w[e] > v1) { v1 = w[e]; i1 = e; } }

#pragma unroll
  for (int e = 0; e < EE; ++e)
    w_masked[(size_t)b * EE + e] = (e == i0 || e == i1) ? w[e] : 0.0f;

  int p0 = atomicAdd(&cnts[i0], 1);
  tokList[i0 * BB + p0] = b; wtList[i0 * BB + p0] = w[i0];
  int p1 = atomicAdd(&cnts[i1], 1);
  tokList[i1 * BB + p1] = b; wtList[i1 * BB + p1] = w[i1];
}

// ---------------------------------------------------------------------------
// Kernel 2: aux loss — deterministic fixed-tree reduction over w_masked
// ---------------------------------------------------------------------------
__global__ __launch_bounds__(256) void aux_kernel(const float* __restrict__ w_masked,
                                                  float* __restrict__ out_aux) {
  __shared__ float part[256];
  const int tid = threadIdx.x;
  const int e = tid & 15;
  const int slice = tid >> 4;  // 16 slices per expert
  float s = 0.f;
  for (int b = slice; b < BB; b += 16) s += w_masked[(size_t)b * EE + e];
  part[tid] = s;
  __syncthreads();
  for (int step = 8; step >= 1; step >>= 1) {
    if (slice < step) part[tid] += part[tid + step * 16];
    __syncthreads();
  }
  if (tid == 0) {
    float c[EE], total = 0.f;
    for (int i = 0; i < EE; ++i) { c[i] = part[i]; total += c[i]; }
    float mean = total / (float)EE;
    float var = 0.f;
    for (int i = 0; i < EE; ++i) { float d = c[i] - mean; var += d * d; }
    var /= (float)(EE - 1);
    float entropy = 0.f;
    for (int i = 0; i < EE; ++i) { float p = c[i] / total; entropy -= p * logf(p + 1e-12f); }
    out_aux[0] = 0.5f * (sqrtf(var) + entropy);
  }
}

// ---------------------------------------------------------------------------
// Kernel 3: per-(expert, 32-token tile) fused MLP via V_WMMA_F32_16X16X4_F32.
// Each B fragment (from L2-resident W1/W2) feeds 2 WMMAs (two M-tiles):
// 16 flops per L2 byte, so the fp32 matrix pipe is the limiter, not L2.
// ---------------------------------------------------------------------------
__global__ __launch_bounds__(256) void moe_kernel(
    const float* __restrict__ h, const float* __restrict__ ce,
    const float* __restrict__ W1, const float* __restrict__ b1,
    const float* __restrict__ W2, const float* __restrict__ b2,
    const int* __restrict__ cnts, const int* __restrict__ tokList,
    const float* __restrict__ wtList, float* __restrict__ out) {
  const int e = blockIdx.y;
  const int tile = blockIdx.x;
  const int n = cnts[e];
  const int base = tile * MT;
  if (base >= n) return;

  const int tid = threadIdx.x;
  const int wave = tid >> 5;
  const int lane = tid & 31;
  const int half = lane >> 4;     // selects K-pair of the 16x16x4 fragment
  const int lm = lane & 15;       // M (A) / N (B,C,D) index

  __shared__ float xs[MT * XS];   // x tile   [32][320] (+pad)  ~41.5 KB
  __shared__ float hs[MT * HS];   // hid chunk [32][128] (+pad) ~16.9 KB
  __shared__ int toks[MT];
  __shared__ float wts[MT];

  if (tid < MT) {
    int idx = base + tid;
    if (idx < n) { toks[tid] = tokList[e * BB + idx]; wts[tid] = wtList[e * BB + idx]; }
    else         { toks[tid] = tokList[e * BB + base]; wts[tid] = 0.0f; }
  }
  __syncthreads();

  for (int i = tid; i < MT * IN; i += 256) {
    int m = i / IN, col = i % IN;
    int t = toks[m];
    float v = (col < HH) ? h[(size_t)t * HH + col] : ce[(size_t)t * CC + (col - HH)];
    xs[m * XS + col] = v;
  }
  __syncthreads();

  const float* W1e = W1 + (size_t)e * O4 * IN;
  const float* b1e = b1 + (size_t)e * O4;
  const float* W2e = W2 + (size_t)e * HH * O4;
  const float* b2e = b2 + (size_t)e * HH;

  // persistent GEMM2 accumulators: [d-tile j][M-tile mt]
  v8f macc[2][2];
#pragma unroll
  for (int j = 0; j < 2; ++j) {
    float bv = b2e[(wave * 2 + j) * 16 + lm];
#pragma unroll
    for (int mt = 0; mt < 2; ++mt)
#pragma unroll
      for (int r = 0; r < 8; ++r) macc[j][mt][r] = bv;
  }

  for (int chunk = 0; chunk < O4 / CH; ++chunk) {  // 8 chunks of 128 cols
    // ---- GEMM1: hid[:, chunk*128 .. +128] = gelu(x @ W1^T + b1) ----
    {
      const int nbg = chunk * CH + wave * 16;  // this wave's output column base
      float bv = b1e[nbg + lm];
      v8f acc0, acc1;
#pragma unroll
      for (int r = 0; r < 8; ++r) { acc0[r] = bv; acc1[r] = bv; }
      const float* x0   = xs + lm * XS + 2 * half;              // rows 0..15
      const float* x1   = xs + (16 + lm) * XS + 2 * half;       // rows 16..31
      const float* wrow = W1e + (size_t)(nbg + lm) * IN + 2 * half;
#pragma unroll 4
      for (int kb = 0; kb < IN; kb += 4) {
        v2f bf = *(const v2f*)(wrow + kb);   // shared across both M-tiles
        v2f a0 = *(const v2f*)(x0 + kb);
        v2f a1 = *(const v2f*)(x1 + kb);
        acc0 = wmma_f32(a0, bf, acc0);
        acc1 = wmma_f32(a1, bf, acc1);
      }
      const int m0 = half * 8;
      const int col = wave * 16 + lm;  // chunk-local column
#pragma unroll
      for (int r = 0; r < 8; ++r) {
        float v = acc0[r];
        v = 0.5f * v * (1.0f + erff(v * 0.70710678118654752f));  // exact GELU
        hs[(m0 + r) * HS + col] = v;
        float u = acc1[r];
        u = 0.5f * u * (1.0f + erff(u * 0.70710678118654752f));
        hs[(16 + m0 + r) * HS + col] = u;
      }
    }
    __syncthreads();

    // ---- GEMM2 partial: moe += hid_chunk @ W2_chunk^T ----
    {
      const int d0 = (wave * 2 + 0) * 16;
      const int d1 = (wave * 2 + 1) * 16;
      const float* h0 = hs + lm * HS + 2 * half;
      const float* h1 = hs + (16 + lm) * HS + 2 * half;
      const float* w0 = W2e + (size_t)(d0 + lm) * O4 + chunk * CH + 2 * half;
      const float* w1 = W2e + (size_t)(d1 + lm) * O4 + chunk * CH + 2 * half;
      v8f a00 = macc[0][0], a01 = macc[0][1], a10 = macc[1][0], a11 = macc[1][1];
#pragma unroll 4
      for (int kb = 0; kb < CH; kb += 4) {
        v2f af0 = *(const v2f*)(h0 + kb);    // A rows 0..15
        v2f af1 = *(const v2f*)(h1 + kb);    // A rows 16..31
        v2f bf0 = *(const v2f*)(w0 + kb);    // B for d-tile 0
        v2f bf1 = *(const v2f*)(w1 + kb);    // B for d-tile 1
        a00 = wmma_f32(af0, bf0, a00);
        a01 = wmma_f32(af1, bf0, a01);
        a10 = wmma_f32(af0, bf1, a10);
        a11 = wmma_f32(af1, bf1, a11);
      }
      macc[0][0] = a00; macc[0][1] = a01; macc[1][0] = a10; macc[1][1] = a11;
    }
    __syncthreads();
  }

  // ---- epilogue: weight-scale and combine (exactly 2 contributions/element) ----
#pragma unroll
  for (int j = 0; j < 2; ++j) {
    const int db = (wave * 2 + j) * 16;
#pragma unroll
    for (int mt = 0; mt < 2; ++mt) {
#pragma unroll
      for (int r = 0; r < 8; ++r) {
        const int m = mt * 16 + half * 8 + r;
        if (base + m < n) {
          float v = macc[j][mt][r] * wts[m];
          atomicAdd(&out[(size_t)toks[m] * HH + db + lm], v);
        }
      }
    }
  }
}

// ---------------------------------------------------------------------------
extern "C" void kernel_launch(void* const* d_in, const int* in_sizes, int n_in,
                              void* d_out, int out_size, void* d_ws, size_t ws_size,
                              hipStream_t stream) {
  const float* h      = (const float*)d_in[0];
  const float* ce     = (const float*)d_in[1];
  const float* anchor = (const float*)d_in[2];
  const float* W1     = (const float*)d_in[3];
  const float* b1     = (const float*)d_in[4];
  const float* W2     = (const float*)d_in[5];
  const float* b2     = (const float*)d_in[6];
  const float* gumbel = (const float*)d_in[7];
  float* out = (float*)d_out;

  char* wsb = (char*)d_ws;
  float* w_masked = (float*)wsb;                              // B*E floats = 1 MB
  int*   cnts     = (int*)(wsb + (size_t)BB * EE * 4);        // E ints
  int*   tokList  = (int*)(wsb + (size_t)BB * EE * 4 + 256);  // E*B ints = 1 MB
  float* wtList   = (float*)(wsb + (size_t)BB * EE * 4 + 256 + (size_t)EE * BB * 4);

  hipMemsetAsync(out, 0, (size_t)BB * HH * sizeof(float), stream);
  hipMemsetAsync(cnts, 0, EE * sizeof(int), stream);

  router_kernel<<<BB / 256, 256, 0, stream>>>(ce, anchor, gumbel, w_masked,
                                              cnts, tokList, wtList);
  aux_kernel<<<1, 256, 0, stream>>>(w_masked, out + (size_t)BB * HH);

  dim3 grid(BB / MT, EE);  // 512 tiles max per expert x 16 experts (early-exit)
  moe_kernel<<<grid, 256, 0, stream>>>(h, ce, W1, b1, W2, b2,
                                       cnts, tokList, wtList, out);
}